// DepthBranch_42580305772560
// MI455X (gfx1250) — compile-verified
//
#include <hip/hip_runtime.h>
#include <hip/hip_bf16.h>

typedef float v2f __attribute__((ext_vector_type(2)));
typedef float v4f __attribute__((ext_vector_type(4)));
typedef float v8f __attribute__((ext_vector_type(8)));

#define HH_ 256
#define WW_ 320
#define BB_ 2
#define DD_ 48
#define CC_ 32

// ---------------------------------------------------------------------------
// Pass 1: zero the 1 GB output volume with nontemporal 128-bit stores.
// This is the bandwidth-dominant pass (~43 us at 23.3 TB/s); output exceeds
// the 192 MB L2, so NT stores are the right temporal hint.
// ---------------------------------------------------------------------------
__global__ void __launch_bounds__(256) zero_out_kernel(v4f* __restrict__ out, long n4) {
    long i = (long)blockIdx.x * blockDim.x + threadIdx.x;
    long stride = (long)gridDim.x * blockDim.x;
    v4f z = {0.f, 0.f, 0.f, 0.f};
    for (; i < n4; i += stride) {
        __builtin_nontemporal_store(z, out + i);
    }
}

// ---------------------------------------------------------------------------
// Pass 2: conv1 (1 -> 16 channels, 3x3, SAME) + ReLU into d_ws.
// Output layout is channel-last (B,H,W,16) so pass 3's B-fragment loads of
// two adjacent input channels are a single aligned b64 load, and this kernel
// writes its 16 channels as four contiguous b128 stores.
// ---------------------------------------------------------------------------
__global__ void __launch_bounds__(256) conv1_kernel(
    const float* __restrict__ x,    // (B,1,H,W)
    const float* __restrict__ w1,   // (16,1,3,3)
    const float* __restrict__ b1,   // (16,)
    float* __restrict__ feat1)      // (B,H,W,16) channel-last
{
    int idx = blockIdx.x * 256 + threadIdx.x;
    const int total = BB_ * HH_ * WW_;
    if (idx >= total) return;
    int b = idx / (HH_ * WW_);
    int p = idx - b * (HH_ * WW_);
    int h = p / WW_;
    int w = p - h * WW_;

    float in[9];
#pragma unroll
    for (int ky = 0; ky < 3; ++ky) {
#pragma unroll
        for (int kx = 0; kx < 3; ++kx) {
            int hh = h + ky - 1;
            int ww = w + kx - 1;
            float v = 0.f;
            if ((unsigned)hh < (unsigned)HH_ && (unsigned)ww < (unsigned)WW_)
                v = x[((size_t)b * HH_ + hh) * WW_ + ww];
            in[ky * 3 + kx] = v;
        }
    }

    float acc[16];
#pragma unroll
    for (int oc = 0; oc < 16; ++oc) {
        float a = b1[oc];  // uniform index -> scalar loads
#pragma unroll
        for (int t = 0; t < 9; ++t) a = fmaf(w1[oc * 9 + t], in[t], a);
        acc[oc] = fmaxf(a, 0.f);
    }

    float* o = feat1 + ((size_t)b * HH_ * WW_ + (size_t)h * WW_ + w) * 16;
#pragma unroll
    for (int q = 0; q < 4; ++q) {
        v4f v = { acc[4 * q], acc[4 * q + 1], acc[4 * q + 2], acc[4 * q + 3] };
        *(v4f*)(o + 4 * q) = v;
    }
}

// ---------------------------------------------------------------------------
// Pass 3: conv2 (16 -> 32, 3x3, SAME) as implicit GEMM on V_WMMA_F32_16X16X4_F32,
// fused with bias+ReLU, per-pixel argmin over depth hypotheses, and scatter of
// the nonzero 1/48 slice into the output volume.
//
// GEMM view: D[oc, n] = sum_K A[oc, K] * B[K, n]
//   K = t*16 + ic  with t = ky*3+kx  (w2 is reordered into this K layout
//   during the LDS fill), K = 144.  Fully unrolled: all (t, ky, kx, ic)
//   are compile-time constants -> no div/mod in the loop body.
//   n = 16 consecutive pixels along W (one strip per wave);
//   oc split into two 16-row halves -> two v8f accumulators per wave.
//
// Boundary handling is branch-free (clamped address + 0/1 mask multiply) so
// EXEC stays all-1s around the WMMAs and loads pipeline freely.
//
// Fragment layouts per ISA (fp32, wave32):
//   A 16x4:  lane = m + 16*g supplies A[m][k0+2g], A[m][k0+2g+1] in vgpr 0,1
//   B 4x16:  lane = n + 16*g supplies B[k0+2g][n], B[k0+2g+1][n]
//   C/D 16x16: vgpr v, lanes 0-15 -> M=v, lanes 16-31 -> M=v+8
// ---------------------------------------------------------------------------
__global__ void __launch_bounds__(128) conv2_scatter_kernel(
    const float* __restrict__ feat1,  // (B,H,W,16) channel-last
    const float* __restrict__ w2,     // (32,16,3,3) flat: oc*144 + ic*9 + t
    const float* __restrict__ b2,     // (32,)
    const float* __restrict__ depth,  // (B,1,H,W)
    const float* __restrict__ hyp,    // (B,48)
    float* __restrict__ out)          // (B,32,48,H,W)
{
    __shared__ float lw2[CC_ * 144];  // 18 KB, layout [oc][t*16 + ic]
    __shared__ float lb2[CC_];
    __shared__ float lhyp[DD_];

    const int quads_per_row = WW_ / 16 / 4;  // 5 (4 waves per block)

    int blk = blockIdx.x;                    // B*H*5 blocks
    int b   = blk / (HH_ * quads_per_row);
    int rem = blk - b * (HH_ * quads_per_row);
    int h   = rem / quads_per_row;
    int wq  = rem - h * quads_per_row;

    int tid = threadIdx.x;
    // Stage w2 into LDS, transposing (ic,t) -> (t,ic) so K = t*16 + ic.
    for (int i = tid; i < CC_ * 144; i += 128) {
        int oc = i / 144;
        int r  = i - oc * 144;
        int ic = r / 9;
        int t  = r - ic * 9;
        lw2[oc * 144 + t * 16 + ic] = w2[i];
    }
    if (tid < CC_) lb2[tid] = b2[tid];
    if (tid < DD_) lhyp[tid] = hyp[b * DD_ + tid];
    __syncthreads();

    int wave = tid >> 5;
    int lane = tid & 31;
    int m    = lane & 15;   // A row (oc within half) / B column (pixel)
    int g    = lane >> 4;   // K-subgroup select
    int wcol = (wq * 4 + wave) * 16 + m;  // this lane's pixel column

    const float* f1b = feat1 + (size_t)b * HH_ * WW_ * 16;

    v8f c0 = {};  // oc 0..15
    v8f c1 = {};  // oc 16..31

#pragma unroll
    for (int ky = 0; ky < 3; ++ky) {
        int hh = h + ky - 1;
        bool hin = (unsigned)hh < (unsigned)HH_;
        int hc = hin ? hh : 0;
#pragma unroll
        for (int kx = 0; kx < 3; ++kx) {
            const int t = ky * 3 + kx;
            int ww = wcol + kx - 1;
            bool win = (unsigned)ww < (unsigned)WW_;
            int wc = win ? ww : 0;
            float mk = (hin && win) ? 1.f : 0.f;
            const float* base = f1b + ((size_t)hc * WW_ + wc) * 16;
#pragma unroll
            for (int ic0 = 0; ic0 < 16; ic0 += 4) {
                int ic = ic0 + 2 * g;                    // even -> 8B aligned
                v2f bv = *(const v2f*)(base + ic);       // global_load_b64
                v2f bb;
                bb.x = bv.x * mk;
                bb.y = bv.y * mk;
                v2f a0 = *(const v2f*)&lw2[m * 144 + t * 16 + ic];        // ds b64
                v2f a1 = *(const v2f*)&lw2[(m + 16) * 144 + t * 16 + ic]; // ds b64
                // 8 args: (neg_a, A, neg_b, B, c_mod, C, reuse_a, reuse_b)
                c0 = __builtin_amdgcn_wmma_f32_16x16x4_f32(false, a0, false, bb,
                                                           (short)0, c0, false, false);
                c1 = __builtin_amdgcn_wmma_f32_16x16x4_f32(false, a1, false, bb,
                                                           (short)0, c1, false, false);
            }
        }
    }

    // argmin over depth hypotheses for this lane's pixel (first-min tie-break,
    // matching jnp.argmin).
    float x = depth[((size_t)b * HH_ + h) * WW_ + wcol];
    int dstar  = 0;
    float best = fabsf(x - lhyp[0]);
    for (int d = 1; d < DD_; ++d) {
        float df = fabsf(x - lhyp[d]);
        if (df < best) { best = df; dstar = d; }
    }

    // Scatter: lane holds column n=wcol, rows M = v + 8*g of each 16x16 D tile.
    size_t pix = (size_t)h * WW_ + wcol;
#pragma unroll
    for (int v = 0; v < 8; ++v) {
        int oc0 = v + 8 * g;         // 0..15
        int oc1 = oc0 + 16;          // 16..31
        float r0 = fmaxf(c0[v] + lb2[oc0], 0.f);
        float r1 = fmaxf(c1[v] + lb2[oc1], 0.f);
        out[(((size_t)b * CC_ + oc0) * DD_ + dstar) * (size_t)(HH_ * WW_) + pix] = r0;
        out[(((size_t)b * CC_ + oc1) * DD_ + dstar) * (size_t)(HH_ * WW_) + pix] = r1;
    }
}

// ---------------------------------------------------------------------------
extern "C" void kernel_launch(void* const* d_in, const int* in_sizes, int n_in,
                              void* d_out, int out_size, void* d_ws, size_t ws_size,
                              hipStream_t stream) {
    const float* depth = (const float*)d_in[0];  // (B,1,H,W)
    const float* hyp   = (const float*)d_in[1];  // (B,48)
    const float* w1    = (const float*)d_in[2];  // (16,1,3,3)
    const float* b1    = (const float*)d_in[3];  // (16,)
    const float* w2    = (const float*)d_in[4];  // (32,16,3,3)
    const float* b2    = (const float*)d_in[5];  // (32,)
    float* out   = (float*)d_out;
    float* feat1 = (float*)d_ws;                 // (B,H,W,16) = 10.5 MB scratch

    // 1) Zero the whole output volume (bandwidth-dominant pass).
    long n4 = (long)out_size / 4;
    zero_out_kernel<<<16384, 256, 0, stream>>>((v4f*)out, n4);

    // 2) conv1 + ReLU into scratch (channel-last).
    int npix = BB_ * HH_ * WW_;
    conv1_kernel<<<(npix + 255) / 256, 256, 0, stream>>>(depth, w1, b1, feat1);

    // 3) conv2 (WMMA implicit GEMM) + ReLU + argmin + scatter.
    int nblocks = BB_ * HH_ * (WW_ / 16 / 4);    // 2560
    conv2_scatter_kernel<<<nblocks, 128, 0, stream>>>(feat1, w2, b2, depth, hyp, out);
}